// QuantumLayer_73005854097988
// MI455X (gfx1250) — compile-verified
//
#include <hip/hip_runtime.h>
#include <hip/hip_bf16.h>
#include <math.h>

// ---------------------------------------------------------------------------
// QuantumLayer on MI455X (gfx1250):
//   1) build_G : 1024x1024 complex64 Gaussian-gate matrix via recurrence
//   2) gemm    : complex GEMM via V_WMMA_F32_16X16X4_F32 (4 real accum chains)
//                + Kerr phase epilogue
// ---------------------------------------------------------------------------

typedef __attribute__((ext_vector_type(2))) float v2f;
typedef __attribute__((ext_vector_type(8))) float v8f;

struct cpx { float x, y; };

__device__ __forceinline__ cpx cmul(cpx a, cpx b) {
    cpx r; r.x = a.x * b.x - a.y * b.y; r.y = a.x * b.y + a.y * b.x; return r;
}
__device__ __forceinline__ cpx cscale(cpx a, float s) {
    cpx r; r.x = a.x * s; r.y = a.y * s; return r;
}
__device__ __forceinline__ cpx cexpz(cpx z) {
    float e = __expf(z.x);
    float s, c; __sincosf(z.y, &s, &c);
    cpx r; r.x = e * c; r.y = e * s; return r;
}

static constexpr int CUT   = 1024;   // cutoff
static constexpr int BATCH = 8192;
static constexpr int BM    = 64;     // modes per block
static constexpr int BN    = 32;     // batch per block
static constexpr int KB    = 32;     // K chunk

// ---------------------------------------------------------------------------
// Kernel 1: build Gaussian matrix G (planar Re/Im planes in workspace).
// One block, 1024 threads (one per column n). Rows advance with a barrier.
// ---------------------------------------------------------------------------
__global__ __launch_bounds__(1024) void build_G_kernel(
    const float* __restrict__ gamma_p,  // [re, im]
    const float* __restrict__ phi_p,    // [phi]
    const float* __restrict__ zeta_p,   // [re, im]
    float* __restrict__ Gr,             // [CUT*CUT]
    float* __restrict__ Gi)             // [CUT*CUT]
{
    const int n = threadIdx.x;

    // --- scalar prep (redundant per thread, trivial cost) -----------------
    const float gr = gamma_p[0], gi = gamma_p[1];
    const float phi = phi_p[0];
    const float zr = zeta_p[0], zi = zeta_p[1];
    const float r     = sqrtf(zr * zr + zi * zi);
    const float delta = atan2f(zi, zr);
    const float th    = tanhf(r);
    const float ch    = coshf(r);
    const float sech  = 1.0f / ch;

    cpx eid; __sincosf(delta, &eid.y, &eid.x);
    cpx eip; __sincosf(phi,   &eip.y, &eip.x);
    cpx cg;  cg.x = gr; cg.y = -gi;                 // conj(gamma)
    cpx gam; gam.x = gr; gam.y = gi;

    // C = exp(-0.5|g|^2 - 0.5 conj(g)^2 e^{id} tanh r) / sqrt(cosh r)
    cpx cg2 = cmul(cg, cg);
    cpx t   = cscale(cmul(cg2, eid), 0.5f * th);
    cpx ex; ex.x = -0.5f * (gr * gr + gi * gi) - t.x; ex.y = -t.y;
    cpx C = cscale(cexpz(ex), 1.0f / sqrtf(ch));

    cpx mu0 = cmul(cg, cscale(eid, th));            // conj(g) e^{id} tanh r
    mu0.x += gam.x; mu0.y += gam.y;                 // + gamma
    cpx mu1 = cscale(cmul(cg, eip), -sech);         // -conj(g) e^{ip} sech r
    cpx S00 = cscale(eid, th);
    cpx S01 = cscale(eip, -sech);
    float a2 = 2.0f * phi - delta;                  // S11 = -e^{i(2p-d)} tanh r
    cpx S11; __sincosf(a2, &S11.y, &S11.x);
    S11 = cscale(S11, -th);

    __shared__ cpx buf[2][CUT];                     // two row buffers, 16 KB

    // zero the "m-2" buffer for row 1
    buf[1][n].x = 0.0f; buf[1][n].y = 0.0f;
    __syncthreads();

    // --- row 0: sequential scan on thread 0 --------------------------------
    if (n == 0) {
        cpx g1 = C, g2; g2.x = 0.0f; g2.y = 0.0f;
        buf[0][0] = C; Gr[0] = C.x; Gi[0] = C.y;
        for (int nn = 1; nn < CUT; ++nn) {
            float sqn1 = sqrtf((float)(nn - 1));
            float inv  = 1.0f / sqrtf((float)nn);
            cpx t1 = cmul(mu1, g1);
            cpx t2 = cmul(S11, g2);
            cpx g; g.x = (t1.x - sqn1 * t2.x) * inv;
                   g.y = (t1.y - sqn1 * t2.y) * inv;
            buf[0][nn] = g; Gr[nn] = g.x; Gi[nn] = g.y;
            g2 = g1; g1 = g;
        }
    }
    __syncthreads();

    // --- rows m >= 1: parallel over n, barrier per row ---------------------
    const float sqn  = sqrtf((float)n);
    for (int m = 1; m < CUT; ++m) {
        cpx p   = buf[(m - 1) & 1][n];                       // G[m-1, n]
        cpx pm1; pm1.x = 0.0f; pm1.y = 0.0f;                 // G[m-1, n-1]
        if (n > 0) pm1 = buf[(m - 1) & 1][n - 1];
        cpx p2  = buf[m & 1][n];                             // G[m-2, n]

        float sqm1 = sqrtf((float)(m - 1));
        float inv  = 1.0f / sqrtf((float)m);

        cpx v  = cmul(mu0, p);
        cpx t2 = cmul(S00, p2);
        cpx t3 = cmul(S01, pm1);
        v.x = (v.x - sqm1 * t2.x - sqn * t3.x) * inv;
        v.y = (v.y - sqm1 * t2.y - sqn * t3.y) * inv;

        buf[m & 1][n] = v;                                   // own slot only
        Gr[m * CUT + n] = v.x;
        Gi[m * CUT + n] = v.y;
        __syncthreads();
    }
}

// ---------------------------------------------------------------------------
// Kernel 2: complex GEMM out[b,m] = sum_n G[m,n] * state[b,n], then Kerr
// phase. 256 threads / 8 waves; each wave owns a 16x16 (mode x batch) tile.
// ---------------------------------------------------------------------------
__global__ __launch_bounds__(256) void qlayer_gemm_kernel(
    const float* __restrict__ state,    // [BATCH][CUT] complex interleaved
    const float* __restrict__ Gr,       // [CUT][CUT]
    const float* __restrict__ Gi,       // [CUT][CUT]
    const float* __restrict__ kappa_p,  // [kappa]
    float* __restrict__ out)            // [BATCH][CUT] complex interleaved
{
    __shared__ float Agr[BM][KB + 1];   // G tile, Re  (pad -> conflict-free)
    __shared__ float Agi[BM][KB + 1];   // G tile, Im
    __shared__ float Bsr[KB][BN + 1];   // state tile, Re  [fock_k][batch]
    __shared__ float Bsi[KB][BN + 1];   // state tile, Im

    const int tid  = threadIdx.x;
    const int lane = tid & 31;
    const int wave = tid >> 5;          // 8 waves
    const int wm   = wave & 3;          // mode sub-tile (4)
    const int wn   = wave >> 2;         // batch sub-tile (2)

    const int mb = blockIdx.x * BM;     // mode base
    const int bb = blockIdx.y * BN;     // batch base

    v8f accRR = {};                     // Gr*Sr
    v8f accII = {};                     // Gi*Si
    v8f accRI = {};                     // Gr*Si
    v8f accIR = {};                     // Gi*Sr

    const int halfk = (lane >> 4) * 2;  // lanes 16-31 hold K+2,K+3
    const int mlF   = (wm << 4) + (lane & 15);   // fragment mode row
    const int nlF   = (wn << 4) + (lane & 15);   // fragment batch col

    for (int kc = 0; kc < CUT; kc += KB) {
        // stage G tile: 64x32 per plane, 8 elems/thread/plane
        #pragma unroll
        for (int i = 0; i < 8; ++i) {
            int idx = tid + i * 256;
            int ml = idx >> 5, kl = idx & 31;
            size_t g = (size_t)(mb + ml) * CUT + (kc + kl);
            Agr[ml][kl] = Gr[g];
            Agi[ml][kl] = Gi[g];
        }
        // stage state tile: 32(batch) x 32(fock) complex, deinterleave
        #pragma unroll
        for (int i = 0; i < 4; ++i) {
            int idx = tid + i * 256;
            int bl = idx >> 5, kf = idx & 31;
            const float* sp = state + ((size_t)(bb + bl) * CUT + (kc + kf)) * 2;
            Bsr[kf][bl] = sp[0];
            Bsi[kf][bl] = sp[1];
        }
        // prefetch next G chunk while this one computes
        if (kc + KB < CUT) {
            __builtin_prefetch(&Gr[(size_t)(mb + (tid >> 2)) * CUT + kc + KB], 0, 1);
            __builtin_prefetch(&Gi[(size_t)(mb + (tid >> 2)) * CUT + kc + KB], 0, 1);
        }
        __syncthreads();

        #pragma unroll
        for (int kk = 0; kk < KB; kk += 4) {
            // A fragment: A[m][k] -> lane=(k>=2?16:0)+m, vgpr=k&1
            v2f ar, ai, br, bi;
            ar.x = Agr[mlF][kk + halfk];     ar.y = Agr[mlF][kk + halfk + 1];
            ai.x = Agi[mlF][kk + halfk];     ai.y = Agi[mlF][kk + halfk + 1];
            // B fragment: B[k][n] -> lane=(k>=2?16:0)+n, vgpr=k&1
            br.x = Bsr[kk + halfk][nlF];     br.y = Bsr[kk + halfk + 1][nlF];
            bi.x = Bsi[kk + halfk][nlF];     bi.y = Bsi[kk + halfk + 1][nlF];

            accRR = __builtin_amdgcn_wmma_f32_16x16x4_f32(
                        false, ar, false, br, (short)0, accRR, false, false);
            accII = __builtin_amdgcn_wmma_f32_16x16x4_f32(
                        false, ai, false, bi, (short)0, accII, false, false);
            accRI = __builtin_amdgcn_wmma_f32_16x16x4_f32(
                        false, ar, false, bi, (short)0, accRI, false, false);
            accIR = __builtin_amdgcn_wmma_f32_16x16x4_f32(
                        false, ai, false, br, (short)0, accIR, false, false);
        }
        __syncthreads();
    }

    // epilogue: combine complex parts, apply Kerr phase, store interleaved
    const float kappa = kappa_p[0];
    #pragma unroll
    for (int v = 0; v < 8; ++v) {
        int m = mb + (wm << 4) + v + ((lane >> 4) << 3);   // D: M = v + 8*(lane/16)
        int b = bb + (wn << 4) + (lane & 15);              // D: N = lane%16
        float re = accRR[v] - accII[v];
        float im = accRI[v] + accIR[v];
        float ang = kappa * (float)m * (float)m;
        float sn, cs; __sincosf(ang, &sn, &cs);
        float* op = out + ((size_t)b * CUT + m) * 2;
        op[0] = re * cs - im * sn;
        op[1] = re * sn + im * cs;
    }
}

// ---------------------------------------------------------------------------
// Launch
// ---------------------------------------------------------------------------
extern "C" void kernel_launch(void* const* d_in, const int* in_sizes, int n_in,
                              void* d_out, int out_size, void* d_ws, size_t ws_size,
                              hipStream_t stream) {
    const float* state   = (const float*)d_in[0];  // complex64 interleaved
    const float* gamma_p = (const float*)d_in[1];  // complex64 scalar
    const float* phi_p   = (const float*)d_in[2];  // float32 scalar
    const float* zeta_p  = (const float*)d_in[3];  // complex64 scalar
    const float* kappa_p = (const float*)d_in[4];  // float32 scalar
    (void)in_sizes; (void)n_in; (void)out_size; (void)ws_size;

    float* Gr = (float*)d_ws;                      // 4 MB
    float* Gi = Gr + (size_t)CUT * CUT;            // 4 MB

    build_G_kernel<<<1, CUT, 0, stream>>>(gamma_p, phi_p, zeta_p, Gr, Gi);

    dim3 grid(CUT / BM, BATCH / BN);               // 16 x 256 blocks
    qlayer_gemm_kernel<<<grid, 256, 0, stream>>>(state, Gr, Gi, kappa_p,
                                                 (float*)d_out);
}